// Stgcn_81123342287290
// MI455X (gfx1250) — compile-verified
//
#include <hip/hip_runtime.h>

#define NB   8
#define LL   2048
#define VV   25
#define PP   3
#define CIN0 3
#define C1   64
#define C2   64
#define C3   128
#define NCC  60
#define TK   9

typedef __attribute__((ext_vector_type(2))) float v2f;
typedef __attribute__((ext_vector_type(8))) float v8f;

__device__ __forceinline__ v8f wmma_f32_4(v2f a, v2f b, v8f c) {
    return __builtin_amdgcn_wmma_f32_16x16x4_f32(false, a, false, b, (short)0, c, false, false);
}

// ---------------------------------------------------------------------------
// Fused channel-GEMM + graph contraction per (n, l).
// e[c,w] = sum_{p,ci} W[p*CO+c, ci] * (x_l @ (A*M)_p)[ci, w] + bias-fold
// The vertex contraction xa = x_l @ AM_p is itself done with WMMA when
// Cin is a multiple of 16 (layers 2/3); VALU fallback for Cin=3 (layer 1).
// ---------------------------------------------------------------------------
template<int CI, int CO>
__global__ __launch_bounds__(128)
void layer_gemm_graph(const float* __restrict__ xin,   // [N][CI][L][V]
                      const float* __restrict__ Wm,    // [P*CO][CI]
                      const float* __restrict__ bv,    // [P*CO]
                      const float* __restrict__ Aadj,  // [P][V][V]
                      const float* __restrict__ Mmask, // [P][V][V]
                      float* __restrict__ eout)        // [N][CO][L][V]
{
    constexpr int KDIM = PP * CI;
    constexpr int KP   = (KDIM + 3) & ~3;

    __shared__ float sAM[PP * VV * VV];   // adjacency * mask
    __shared__ float sCS[PP][32];         // column sums (bias fold), zero-padded
    __shared__ float sX[CI][VV];          // x slice for this (n,l)
    __shared__ float sXA[KP][32];         // xa = x_l @ AM_p, K-rows, 32 padded cols

    const int blk = blockIdx.x;
    const int n   = blk / LL;
    const int l   = blk % LL;
    const int tid = threadIdx.x;

    for (int i = tid; i < PP * VV * VV; i += 128)
        sAM[i] = Aadj[i] * Mmask[i];
    for (int i = tid; i < CI * VV; i += 128) {
        int ci = i / VV, v = i % VV;
        sX[ci][v] = xin[((n * CI + ci) * LL + l) * VV + v];
    }
    __syncthreads();

    for (int i = tid; i < PP * 32; i += 128) {
        int p = i >> 5, w = i & 31;
        float s = 0.f;
        if (w < VV)
            for (int v = 0; v < VV; ++v) s += sAM[(p * VV + v) * VV + w];
        sCS[p][w] = s;
    }

    const int lane = tid & 31;
    const int wave = tid >> 5;
    const int lcol = lane & 15;
    const int lhi  = lane >> 4;

    if constexpr ((CI % 16) == 0) {
        // --- xa via WMMA: per p, [CI x 25] @ [25 x 25], K padded to 28 ---
        constexpr int MT = CI / 16;             // m-tiles per partition
        constexpr int XJOBS = PP * MT * 2;      // x 2 n-tiles
        for (int job = wave; job < XJOBS; job += 4) {
            const int p  = job / (MT * 2);
            const int r2 = job % (MT * 2);
            const int m0 = (r2 >> 1) * 16;
            const int n0 = (r2 & 1) * 16;
            v8f acc = {};
            for (int kk = 0; kk < 28; kk += 4) {
                const int ka = kk + 2 * lhi;
                v2f a, b;
                const int col = n0 + lcol;
                a.x = (ka     < VV) ? sX[m0 + lcol][ka]     : 0.f;
                a.y = (ka + 1 < VV) ? sX[m0 + lcol][ka + 1] : 0.f;
                b.x = (ka     < VV && col < VV) ? sAM[(p * VV + ka    ) * VV + col] : 0.f;
                b.y = (ka + 1 < VV && col < VV) ? sAM[(p * VV + ka + 1) * VV + col] : 0.f;
                acc = wmma_f32_4(a, b, acc);
            }
            #pragma unroll
            for (int r = 0; r < 8; ++r) {
                int row = m0 + r + 8 * lhi;       // ci
                sXA[p * CI + row][n0 + lcol] = acc[r];  // cols >= 25 are exact zeros
            }
        }
    } else {
        // --- VALU fallback for tiny Cin (layer 1: KDIM=9, KP=12) ---
        for (int i = tid; i < KP * 32; i += 128) {
            int row = i >> 5, w = i & 31;
            float s = 0.f;
            if (row < KDIM && w < VV) {
                int p = row / CI, ci = row % CI;
                for (int v = 0; v < VV; ++v) s += sX[ci][v] * sAM[(p * VV + v) * VV + w];
            }
            sXA[row][w] = s;
        }
    }
    __syncthreads();

    constexpr int NJOBS = (CO / 16) * 2;  // m-tiles x 2 n-tiles (25 -> 32)
    for (int job = wave; job < NJOBS; job += 4) {
        const int m0 = (job >> 1) * 16;
        const int n0 = (job & 1) * 16;
        // C init: bias folded through adjacency column sums
        v8f acc;
        #pragma unroll
        for (int r = 0; r < 8; ++r) {
            int row = m0 + r + 8 * lhi;
            float s = 0.f;
            #pragma unroll
            for (int p = 0; p < PP; ++p) s += bv[p * CO + row] * sCS[p][n0 + lcol];
            acc[r] = s;
        }
        for (int kk = 0; kk < KP; kk += 4) {
            const int ka = kk + 2 * lhi;
            v2f a, b;
            const int mrow = m0 + lcol;
            a.x = (ka     < KDIM) ? Wm[(( ka      / CI) * CO + mrow) * CI + ( ka      % CI)] : 0.f;
            a.y = (ka + 1 < KDIM) ? Wm[(((ka + 1) / CI) * CO + mrow) * CI + ((ka + 1) % CI)] : 0.f;
            b.x = sXA[ka][n0 + lcol];
            b.y = sXA[ka + 1][n0 + lcol];
            acc = wmma_f32_4(a, b, acc);
        }
        #pragma unroll
        for (int r = 0; r < 8; ++r) {
            int row = m0 + r + 8 * lhi;
            int w   = n0 + lcol;
            if (w < VV) eout[((n * CO + row) * LL + l) * VV + w] = acc[r];
        }
    }
}

// ---------------------------------------------------------------------------
// Causal 9-tap temporal sum + residual + ReLU (e is mostly L2-resident)
// ---------------------------------------------------------------------------
template<int CO>
__global__ __launch_bounds__(256)
void temporal_res_relu(const float* __restrict__ e,
                       const float* __restrict__ res,
                       float* __restrict__ y)
{
    const long long total = (long long)NB * CO * LL * VV;
    long long i = (long long)blockIdx.x * 256 + threadIdx.x;
    if (i >= total) return;
    const int l = (int)((i / VV) % LL);
    float s = 0.f;
    #pragma unroll
    for (int k = 0; k < TK; ++k)
        if (l - k >= 0) s += e[i - (long long)k * VV];
    float r = s + res[i];
    y[i] = r > 0.f ? r : 0.f;
}

// ---------------------------------------------------------------------------
// Layer-1 residual: 3-channel GEMM + BN affine (K=3, trivial -> VALU)
// ---------------------------------------------------------------------------
__global__ __launch_bounds__(256)
void res_bn_cin3(const float* __restrict__ x, const float* __restrict__ Wr,
                 const float* __restrict__ br, const float* __restrict__ g,
                 const float* __restrict__ be, const float* __restrict__ mm,
                 const float* __restrict__ vv, float* __restrict__ out)
{
    const long long total = (long long)NB * C1 * LL * VV;
    long long i = (long long)blockIdx.x * 256 + threadIdx.x;
    if (i >= total) return;
    const int lv = (int)(i % (LL * VV));
    long long t  = i / (LL * VV);
    const int o  = (int)(t % C1);
    const int n  = (int)(t / C1);
    const float* xb = x + (long long)n * CIN0 * LL * VV + lv;
    float z = xb[0] * Wr[o * CIN0 + 0]
            + xb[(long long)LL * VV] * Wr[o * CIN0 + 1]
            + xb[2LL * LL * VV] * Wr[o * CIN0 + 2] + br[o];
    float inv = rsqrtf(vv[o] + 1e-5f);
    out[i] = g[o] * (z - mm[o]) * inv + be[o];
}

// ---------------------------------------------------------------------------
// Layer-3 residual: [128x64] GEMM over 51200 positions per n + BN (WMMA)
// ---------------------------------------------------------------------------
__global__ __launch_bounds__(128)
void res_bn_wmma(const float* __restrict__ xin,  // y2: [N][64][L][V]
                 const float* __restrict__ Wr,   // [128][64]
                 const float* __restrict__ br, const float* __restrict__ g,
                 const float* __restrict__ be, const float* __restrict__ mm,
                 const float* __restrict__ vv,
                 float* __restrict__ out)        // [N][128][L][V]
{
    __shared__ float sB[64][65];
    __shared__ float sScale[128], sOff[128];
    const int LV  = LL * VV;           // 51200
    const int blk = blockIdx.x;
    const int n   = blk / (LV / 64);
    const int j0  = (blk % (LV / 64)) * 64;
    const int tid = threadIdx.x;

    {
        int o = tid;                    // 128 threads == 128 rows
        float inv = rsqrtf(vv[o] + 1e-5f) * g[o];
        sScale[o] = inv;
        sOff[o]   = be[o] + (br[o] - mm[o]) * inv;
    }
    for (int i = tid; i < 64 * 64; i += 128) {
        int k = i >> 6, jj = i & 63;
        sB[k][jj] = xin[(long long)(n * 64 + k) * LV + j0 + jj];
    }
    __syncthreads();

    const int lane = tid & 31, wave = tid >> 5;
    const int lcol = lane & 15, lhi = lane >> 4;
    for (int job = wave; job < 32; job += 4) {
        const int m0 = (job >> 2) * 16;  // 8 m-tiles
        const int n0 = (job & 3) * 16;   // 4 n-tiles
        v8f acc = {};
        for (int kk = 0; kk < 64; kk += 4) {
            const int ka = kk + 2 * lhi;
            v2f a, b;
            a.x = Wr[(m0 + lcol) * 64 + ka];
            a.y = Wr[(m0 + lcol) * 64 + ka + 1];
            b.x = sB[ka][n0 + lcol];
            b.y = sB[ka + 1][n0 + lcol];
            acc = wmma_f32_4(a, b, acc);
        }
        #pragma unroll
        for (int r = 0; r < 8; ++r) {
            int row = m0 + r + 8 * lhi;
            out[(long long)(n * 128 + row) * LV + j0 + n0 + lcol]
                = acc[r] * sScale[row] + sOff[row];
        }
    }
}

// ---------------------------------------------------------------------------
// V-mean pool + FC (60x128 GEMM over 16 l's per block, WMMA)
// ---------------------------------------------------------------------------
__global__ __launch_bounds__(128)
void pool_fc(const float* __restrict__ y3,   // [N][128][L][V]
             const float* __restrict__ Wfc,  // [60][128]
             const float* __restrict__ bfc,  // [60]
             float* __restrict__ out)        // [N][60][L]
{
    __shared__ float sP[128][17];
    const int blk = blockIdx.x;              // N * (L/16)
    const int n   = blk / (LL / 16);
    const int l0  = (blk % (LL / 16)) * 16;
    const int tid = threadIdx.x;

    {
        int c = tid;                          // one channel per thread
        for (int lt = 0; lt < 16; ++lt) {
            const float* pp = y3 + ((long long)(n * 128 + c) * LL + l0 + lt) * VV;
            float s = 0.f;
            #pragma unroll
            for (int w = 0; w < VV; ++w) s += pp[w];
            sP[c][lt] = s * (1.0f / VV);
        }
    }
    __syncthreads();

    const int lane = tid & 31, wave = tid >> 5;
    const int lcol = lane & 15, lhi = lane >> 4;
    const int m0 = wave * 16;                 // 4 m-tiles (60 -> 64 padded)
    v8f acc;
    #pragma unroll
    for (int r = 0; r < 8; ++r) {
        int row = m0 + r + 8 * lhi;
        acc[r] = (row < NCC) ? bfc[row] : 0.f;
    }
    for (int kk = 0; kk < 128; kk += 4) {
        const int ka = kk + 2 * lhi;
        const int row = m0 + lcol;
        v2f a, b;
        a.x = (row < NCC) ? Wfc[row * 128 + ka] : 0.f;
        a.y = (row < NCC) ? Wfc[row * 128 + ka + 1] : 0.f;
        b.x = sP[ka][lcol];
        b.y = sP[ka + 1][lcol];
        acc = wmma_f32_4(a, b, acc);
    }
    #pragma unroll
    for (int r = 0; r < 8; ++r) {
        int row = m0 + r + 8 * lhi;
        if (row < NCC) out[(long long)(n * NCC + row) * LL + l0 + lcol] = acc[r];
    }
}

// ---------------------------------------------------------------------------
extern "C" void kernel_launch(void* const* d_in, const int* in_sizes, int n_in,
                              void* d_out, int out_size, void* d_ws, size_t ws_size,
                              hipStream_t stream) {
    (void)in_sizes; (void)n_in; (void)out_size; (void)ws_size;
    const float* x   = (const float*)d_in[0];
    const float* Aj  = (const float*)d_in[1];
    const float* W1  = (const float*)d_in[2];
    const float* b1  = (const float*)d_in[3];
    const float* M1  = (const float*)d_in[4];
    const float* Wr1 = (const float*)d_in[5];
    const float* br1 = (const float*)d_in[6];
    const float* g1  = (const float*)d_in[7];
    const float* be1 = (const float*)d_in[8];
    const float* m1  = (const float*)d_in[9];
    const float* v1  = (const float*)d_in[10];
    const float* W2  = (const float*)d_in[11];
    const float* b2  = (const float*)d_in[12];
    const float* M2  = (const float*)d_in[13];
    const float* W3  = (const float*)d_in[14];
    const float* b3  = (const float*)d_in[15];
    const float* M3  = (const float*)d_in[16];
    const float* Wr3 = (const float*)d_in[17];
    const float* br3 = (const float*)d_in[18];
    const float* g3  = (const float*)d_in[19];
    const float* be3 = (const float*)d_in[20];
    const float* m3  = (const float*)d_in[21];
    const float* v3  = (const float*)d_in[22];
    const float* Wfc = (const float*)d_in[23];
    const float* bfc = (const float*)d_in[24];

    float* ws = (float*)d_ws;
    const size_t szBig = (size_t)NB * C3 * LL * VV;  // 52,428,800
    const size_t szSm  = (size_t)NB * C1 * LL * VV;  // 26,214,400
    float* bufE  = ws;                  // e (per layer)
    float* bufR  = ws + szBig;          // res1 / res3
    float* bufY1 = ws + 2 * szBig;      // y1 (later reused for y3: y1+y2 region)
    float* bufY2 = bufY1 + szSm;        // y2
    float* bufY3 = bufY1;               // y3 overwrites dead y1+y2 (szBig)

    const int tblkS = (int)(((long long)NB * C1 * LL * VV + 255) / 256);
    const int tblkB = (int)(((long long)NB * C3 * LL * VV + 255) / 256);

    // residual 1 (BN of 3->64 GEMM)
    res_bn_cin3<<<tblkS, 256, 0, stream>>>(x, Wr1, br1, g1, be1, m1, v1, bufR);
    // layer 1
    layer_gemm_graph<CIN0, C1><<<NB * LL, 128, 0, stream>>>(x, W1, b1, Aj, M1, bufE);
    temporal_res_relu<C1><<<tblkS, 256, 0, stream>>>(bufE, bufR, bufY1);
    // layer 2 (residual = y1)
    layer_gemm_graph<C1, C2><<<NB * LL, 128, 0, stream>>>(bufY1, W2, b2, Aj, M2, bufE);
    temporal_res_relu<C2><<<tblkS, 256, 0, stream>>>(bufE, bufY1, bufY2);
    // residual 3 (BN of 64->128 GEMM, WMMA)
    res_bn_wmma<<<NB * (LL * VV / 64), 128, 0, stream>>>(bufY2, Wr3, br3, g3, be3, m3, v3, bufR);
    // layer 3
    layer_gemm_graph<C2, C3><<<NB * LL, 128, 0, stream>>>(bufY2, W3, b3, Aj, M3, bufE);
    temporal_res_relu<C3><<<tblkB, 256, 0, stream>>>(bufE, bufR, bufY3);
    // pool + FC
    pool_fc<<<NB * (LL / 16), 128, 0, stream>>>(bufY3, Wfc, bfc, (float*)d_out);
}